// TreeMessageEncoder_76836964925860
// MI455X (gfx1250) — compile-verified
//
#include <hip/hip_runtime.h>
#include <hip/hip_bf16.h>
#include <math.h>

// ---------------------------------------------------------------------------
// TreeMessageEncoder for MI455X (gfx1250)
//  - all GEMMs via V_WMMA_F32_16X16X4_F32 (fp32 in / fp32 accumulate)
//  - LDS staging via GLOBAL_LOAD_ASYNC_TO_LDS_B128 + s_wait_asynccnt
//    (double buffered), guarded sync fallback for edge tiles / odd K
// ---------------------------------------------------------------------------

typedef float v2f __attribute__((ext_vector_type(2)));
typedef float v8f __attribute__((ext_vector_type(8)));
typedef int   v4i __attribute__((vector_size(16)));
typedef __attribute__((address_space(1))) v4i v4i_glob;   // printed as "__device__"
typedef __attribute__((address_space(3))) v4i v4i_lds;    // printed as "__shared__"

#define B_   8
#define N_   341
#define S_   32
#define H_   512
#define D_   256
#define NH_  4
#define DH_  64
#define FF_  2048
#define L_   2
#define BN_  (B_ * N_)

#define TM 128
#define TN 64
#define TK 32

enum { ACT_NONE = 0, ACT_RELU = 1, ACT_GELU = 2 };

#if defined(__has_builtin)
#if __has_builtin(__builtin_amdgcn_global_load_async_to_lds_b128) && \
    __has_builtin(__builtin_amdgcn_s_wait_asynccnt)
#define HAVE_ASYNC_LDS 1
#endif
#endif
#ifndef HAVE_ASYNC_LDS
#define HAVE_ASYNC_LDS 0
#endif

#if HAVE_ASYNC_LDS
__device__ __forceinline__ void async_cp16(const float* g, float* l)
{
    // (global src, lds dst, imm offset, imm cpol) -> global_load_async_to_lds_b128
    __builtin_amdgcn_global_load_async_to_lds_b128(
        (v4i_glob*)const_cast<float*>(g), (v4i_lds*)l, 0, 0);
}
#endif

__device__ __forceinline__ v8f wmma4(v2f a, v2f b, v8f c)
{
    return __builtin_amdgcn_wmma_f32_16x16x4_f32(false, a, false, b,
                                                 (short)0, c, false, false);
}

// ---------------------------------------------------------------------------
// Generic batched GEMM:  C = residual + row_mask * act(A @ B + bias)
// A: [batch] M x K row-major, B: [batch] K x N row-major, C: [batch] M x N.
// Block tile 128x64; 8 waves; each wave owns a 2x2 grid of 16x16 tiles
// (4 v_wmma per 4 LDS fragment loads). A staged m-major in LDS so an A
// fragment is one ds_load_b64.
// ---------------------------------------------------------------------------
__global__ __launch_bounds__(256) void gemm_wmma_f32(
    const float* __restrict__ A, const float* __restrict__ Bm,
    const float* __restrict__ bias, const float* __restrict__ residual,
    const float* __restrict__ row_mask, int mask_mod,
    float* __restrict__ C,
    int M, int K, int N,
    long long sA, long long sB, long long sC,
    int act)
{
    __shared__ float Am[2][TM][36];   // [buf][m][k]  (row 144B, 16B aligned)
    __shared__ float Bs[2][TK][68];   // [buf][k][n]  (row 272B, 16B aligned)

    const int tid  = threadIdx.x;
    const int lane = tid & 31;
    const int wave = tid >> 5;
    const int hf   = lane >> 4;          // lane half (0/1)
    const int lm   = lane & 15;
    const int wm   = (wave >> 1) * 32;   // wave m offset in [0,128)
    const int wn   = (wave & 1) * 32;    // wave n offset in [0,64)

    const int mBase = blockIdx.y * TM;
    const int nBase = blockIdx.x * TN;
    const long long bat = blockIdx.z;
    A  += bat * sA;
    Bm += bat * sB;
    C  += bat * sC;
    const float* R = residual ? residual + bat * sC : nullptr;

    v8f acc00 = {0.f,0.f,0.f,0.f,0.f,0.f,0.f,0.f};
    v8f acc01 = acc00, acc10 = acc00, acc11 = acc00;

    const int nk = (K + TK - 1) / TK;

    auto compute = [&](int buf) {
        #pragma unroll
        for (int kk = 0; kk < TK; kk += 4) {
            // A 16x4 frags: lanes 0-15 -> K={kk,kk+1}, lanes 16-31 -> {kk+2,kk+3}
            v2f a0 = *(const v2f*)&Am[buf][wm + lm][kk + 2 * hf];
            v2f a1 = *(const v2f*)&Am[buf][wm + 16 + lm][kk + 2 * hf];
            // B 4x16 frags (row striping mirrors C/D layout)
            v2f b0, b1;
            b0.x = Bs[buf][kk + 2 * hf][wn + lm];
            b0.y = Bs[buf][kk + 1 + 2 * hf][wn + lm];
            b1.x = Bs[buf][kk + 2 * hf][wn + 16 + lm];
            b1.y = Bs[buf][kk + 1 + 2 * hf][wn + 16 + lm];
            acc00 = wmma4(a0, b0, acc00);
            acc01 = wmma4(a0, b1, acc01);
            acc10 = wmma4(a1, b0, acc10);
            acc11 = wmma4(a1, b1, acc11);
        }
    };

#if HAVE_ASYNC_LDS
    const bool full = (mBase + TM <= M) && (nBase + TN <= N) && ((K & (TK - 1)) == 0);
    auto prefetch = [&](int kt, int buf) {
        const int k0 = kt * TK;
        #pragma unroll
        for (int i = 0; i < 4; ++i) {             // A: 128x32 = 1024 x b128
            int e = tid + i * 256;
            int m = e >> 3, seg = (e & 7) * 4;
            async_cp16(&A[(long long)(mBase + m) * K + k0 + seg], &Am[buf][m][seg]);
        }
        #pragma unroll
        for (int i = 0; i < 2; ++i) {             // B: 32x64 = 512 x b128
            int e = tid + i * 256;
            int kk = e >> 4, seg = (e & 15) * 4;
            async_cp16(&Bm[(long long)(k0 + kk) * N + nBase + seg], &Bs[buf][kk][seg]);
        }
    };
    if (full) {
        prefetch(0, 0);
        int cur = 0;
        for (int kt = 0; kt < nk; ++kt) {
            if (kt + 1 < nk) {
                prefetch(kt + 1, cur ^ 1);                 // overlap next tile
                __builtin_amdgcn_s_wait_asynccnt(6);       // tile kt landed
            } else {
                __builtin_amdgcn_s_wait_asynccnt(0);
            }
            __syncthreads();
            compute(cur);
            __syncthreads();                               // free buf before reuse
            cur ^= 1;
        }
    } else
#endif
    {
        for (int kt = 0; kt < nk; ++kt) {
            const int k0 = kt * TK;
            for (int e = tid; e < TM * TK; e += 256) {
                int m = e >> 5, kk = e & 31;
                int gm = mBase + m, gk = k0 + kk;
                Am[0][m][kk] = (gm < M && gk < K) ? A[(long long)gm * K + gk] : 0.f;
            }
            for (int e = tid; e < TK * TN; e += 256) {
                int kk = e >> 6, n = e & 63;
                int gk = k0 + kk, gn = nBase + n;
                Bs[0][kk][n] = (gk < K && gn < N) ? Bm[(long long)gk * N + gn] : 0.f;
            }
            __syncthreads();
            compute(0);
            __syncthreads();
        }
    }

    // Epilogue: C/D layout -> VGPR r holds rows (r, r+8) split across lane halves
    auto store_tile = [&](const v8f& ac, int mi, int ni) {
        #pragma unroll
        for (int r = 0; r < 8; ++r) {
            int gm = mBase + wm + mi * 16 + r + 8 * hf;
            if (gm >= M) continue;
            int gn = nBase + wn + ni * 16 + lm;
            if (gn >= N) continue;
            float v = ac[r] + (bias ? bias[gn] : 0.f);
            if (act == ACT_RELU)      v = fmaxf(v, 0.f);
            else if (act == ACT_GELU) v = 0.5f * v * (1.f + erff(v * 0.70710678118654752f));
            if (row_mask) v *= row_mask[gm % mask_mod];
            if (R) v += R[(long long)gm * N + gn];
            C[(long long)gm * N + gn] = v;
        }
    };
    store_tile(acc00, 0, 0);
    store_tile(acc01, 0, 1);
    store_tile(acc10, 1, 0);
    store_tile(acc11, 1, 1);
}

// ---------------------------------------------------------------------------
// Masked mean-pool over S:  pooled[b,n,:] = sum_s m*x / max(count,1)
// ---------------------------------------------------------------------------
__global__ __launch_bounds__(256) void pool_kernel(
    const float* __restrict__ states, const unsigned char* __restrict__ nmask,
    float* __restrict__ pooled)
{
    const long long row = blockIdx.x;             // b*N + n
    const int t = threadIdx.x;
    const float* base = states + row * (long long)(S_ * H_);
    const unsigned char* mrow = nmask + row * S_;
    float a0 = 0.f, a1 = 0.f;
    int cnt = 0;
    for (int s = 0; s < S_; ++s) {
        if (mrow[s]) {
            ++cnt;
            a0 += base[(long long)s * H_ + t];
            a1 += base[(long long)s * H_ + t + 256];
        }
    }
    float inv = 1.f / (float)(cnt > 0 ? cnt : 1);
    pooled[row * H_ + t]       = a0 * inv;
    pooled[row * H_ + t + 256] = a1 * inv;
}

// x += depth_emb[depth[n]] + pos_emb[n]
__global__ void add_emb_kernel(float* __restrict__ x,
                               const float* __restrict__ de,
                               const float* __restrict__ pe,
                               const int* __restrict__ depth)
{
    long long i = (long long)blockIdx.x * 256 + threadIdx.x;
    if (i >= (long long)BN_ * D_) return;
    int j = (int)(i % D_);
    long long row = i / D_;
    int n = (int)(row % N_);
    x[i] += de[(long long)depth[n] * D_ + j] + pe[(long long)n * D_ + j];
}

// row_mask[n] = (depth[n]==lvl && parent[n]>=0)
__global__ void edge_mask_kernel(float* __restrict__ mk,
                                 const int* __restrict__ depth,
                                 const int* __restrict__ parent, int lvl)
{
    int n = blockIdx.x * 256 + threadIdx.x;
    if (n < N_) mk[n] = (depth[n] == lvl && parent[n] >= 0) ? 1.f : 0.f;
}

// ein[row] = concat(x[parent(row)], x[row])
__global__ void gather_concat_kernel(const float* __restrict__ x,
                                     const int* __restrict__ parent,
                                     float* __restrict__ ein)
{
    long long i = (long long)blockIdx.x * 256 + threadIdx.x;
    if (i >= (long long)BN_ * D_) return;
    int j = (int)(i % D_);
    long long row = i / D_;
    int n = (int)(row % N_);
    long long bbase = row - n;
    int p = parent[n]; if (p < 0) p = 0;
    ein[row * (2 * D_) + j]      = x[(bbase + p) * D_ + j];
    ein[row * (2 * D_) + D_ + j] = x[row * D_ + j];
}

// (B,N,NH,dh) -> dst (B,NH,N,dh) and/or dstT (B,NH,dh,N)
__global__ void pack_heads_kernel(const float* __restrict__ src,
                                  float* __restrict__ dst,
                                  float* __restrict__ dstT)
{
    long long i = (long long)blockIdx.x * 256 + threadIdx.x;
    if (i >= (long long)BN_ * D_) return;
    int d = (int)(i & (DH_ - 1));
    long long r = i >> 6;
    int h = (int)(r & (NH_ - 1)); r >>= 2;
    int n = (int)(r % N_);
    int b = (int)(r / N_);
    float v = src[i];
    long long bh = (long long)(b * NH_ + h);
    if (dst)  dst[(bh * N_ + n) * DH_ + d] = v;
    if (dstT) dstT[(bh * DH_ + d) * N_ + n] = v;
}

// (B,NH,N,dh) -> (B,N,NH,dh)
__global__ void unpack_heads_kernel(const float* __restrict__ srcP,
                                    float* __restrict__ dst)
{
    long long i = (long long)blockIdx.x * 256 + threadIdx.x;
    if (i >= (long long)BN_ * D_) return;
    int d = (int)(i & (DH_ - 1));
    long long r = i >> 6;
    int h = (int)(r & (NH_ - 1)); r >>= 2;
    int n = (int)(r % N_);
    int b = (int)(r / N_);
    dst[i] = srcP[(((long long)(b * NH_ + h)) * N_ + n) * DH_ + d];
}

// scale by 1/sqrt(dh), key-padding mask (-1e9), row softmax over N keys
__global__ __launch_bounds__(128) void softmax_kernel(
    float* __restrict__ scores, const unsigned char* __restrict__ active)
{
    const int row = blockIdx.x;                 // b*NH*N + h*N + q
    const int b = row / (NH_ * N_);
    float* s = scores + (long long)row * N_;
    const unsigned char* am = active + (long long)b * N_;
    const int t = threadIdx.x;

    float v[3];
    float m = -1e30f;
    #pragma unroll
    for (int i = 0; i < 3; ++i) {
        int k = t + i * 128;
        float x = -1e30f;
        if (k < N_) {
            x = s[k] * 0.125f;                  // 1/sqrt(64)
            if (!am[k]) x = -1.0e9f;
        }
        v[i] = x;
        m = fmaxf(m, x);
    }
    __shared__ float red[128];
    red[t] = m; __syncthreads();
    for (int o = 64; o > 0; o >>= 1) { if (t < o) red[t] = fmaxf(red[t], red[t + o]); __syncthreads(); }
    m = red[0]; __syncthreads();

    float sum = 0.f;
    #pragma unroll
    for (int i = 0; i < 3; ++i) {
        int k = t + i * 128;
        if (k < N_) { v[i] = expf(v[i] - m); sum += v[i]; }
    }
    red[t] = sum; __syncthreads();
    for (int o = 64; o > 0; o >>= 1) { if (t < o) red[t] += red[t + o]; __syncthreads(); }
    float inv = 1.f / red[0];
    #pragma unroll
    for (int i = 0; i < 3; ++i) {
        int k = t + i * 128;
        if (k < N_) s[k] = v[i] * inv;
    }
}

// per-row LayerNorm over D=256, in place
__global__ __launch_bounds__(256) void layernorm_kernel(
    float* __restrict__ x, const float* __restrict__ g, const float* __restrict__ b)
{
    const long long row = blockIdx.x;
    const int t = threadIdx.x;
    float* r = x + row * D_;
    float val = r[t];
    __shared__ float red[256];
    red[t] = val; __syncthreads();
    for (int o = 128; o > 0; o >>= 1) { if (t < o) red[t] += red[t + o]; __syncthreads(); }
    float mean = red[0] * (1.f / D_); __syncthreads();
    float d = val - mean;
    red[t] = d * d; __syncthreads();
    for (int o = 128; o > 0; o >>= 1) { if (t < o) red[t] += red[t + o]; __syncthreads(); }
    float var = red[0] * (1.f / D_);
    r[t] = d * (1.f / sqrtf(var + 1e-5f)) * g[t] + b[t];
}

// qh[j] = bq[j] + sum_i agg_query[i] * wq[i,j]   (same for all b)
__global__ void agg_q_kernel(const float* __restrict__ q, const float* __restrict__ wq,
                             const float* __restrict__ bq, float* __restrict__ qh)
{
    int j = threadIdx.x;
    float acc = bq[j];
    for (int i = 0; i < D_; ++i) acc += q[i] * wq[(long long)i * D_ + j];
    qh[j] = acc;
}

// single-query attention per (b,h): scores->softmax->weighted sum of V
__global__ __launch_bounds__(256) void agg_attn_kernel(
    const float* __restrict__ qh, const float* __restrict__ kh,
    const float* __restrict__ vh, const unsigned char* __restrict__ active,
    float* __restrict__ oagg)
{
    const int bh = blockIdx.x;
    const int b = bh / NH_, h = bh % NH_;
    const int t = threadIdx.x;
    __shared__ float sp[N_ + 11];
    __shared__ float red[256];

    const float* qr = qh + h * DH_;
    for (int k = t; k < N_; k += 256) {
        const float* kr = kh + (((long long)(b * N_ + k)) * NH_ + h) * DH_;
        float a = 0.f;
        for (int d = 0; d < DH_; ++d) a += qr[d] * kr[d];
        a *= 0.125f;
        if (!active[b * N_ + k]) a = -1.0e9f;
        sp[k] = a;
    }
    __syncthreads();
    float m = -1e30f;
    for (int k = t; k < N_; k += 256) m = fmaxf(m, sp[k]);
    red[t] = m; __syncthreads();
    for (int o = 128; o > 0; o >>= 1) { if (t < o) red[t] = fmaxf(red[t], red[t + o]); __syncthreads(); }
    m = red[0]; __syncthreads();
    float s = 0.f;
    for (int k = t; k < N_; k += 256) { float e = expf(sp[k] - m); sp[k] = e; s += e; }
    red[t] = s; __syncthreads();
    for (int o = 128; o > 0; o >>= 1) { if (t < o) red[t] += red[t + o]; __syncthreads(); }
    float inv = 1.f / red[0];
    __syncthreads();
    if (t < DH_) {
        float a = 0.f;
        for (int k = 0; k < N_; ++k)
            a += sp[k] * vh[(((long long)(b * N_ + k)) * NH_ + h) * DH_ + t];
        oagg[b * D_ + h * DH_ + t] = a * inv;
    }
}

// ---------------------------------------------------------------------------
// Host side
// ---------------------------------------------------------------------------
static inline void launch_gemm(const float* A, const float* Bm, const float* bias,
                               const float* residual, const float* row_mask, int mask_mod,
                               float* C, int M, int K, int N, int batch,
                               long long sA, long long sB, long long sC,
                               int act, hipStream_t stream)
{
    dim3 grid((N + TN - 1) / TN, (M + TM - 1) / TM, batch);
    gemm_wmma_f32<<<grid, 256, 0, stream>>>(A, Bm, bias, residual, row_mask, mask_mod,
                                            C, M, K, N, sA, sB, sC, act);
}

struct LayerP {
    const float *bk, *bo, *bq, *bv, *ffb1, *ffb2, *ffw1, *ffw2;
    const float *ln1_b, *ln1_g, *ln2_b, *ln2_g, *wk, *wo, *wq, *wv;
};

extern "C" void kernel_launch(void* const* d_in, const int* in_sizes, int n_in,
                              void* d_out, int out_size, void* d_ws, size_t ws_size,
                              hipStream_t stream)
{
    (void)in_sizes; (void)n_in; (void)out_size;
    auto F = [&](int i) { return (const float*)d_in[i]; };

    // JAX pytree flatten order (sorted dict keys):
    const unsigned char* active = (const unsigned char*)d_in[0];
    const float* states   = F(1);
    const int*   depth    = (const int*)d_in[2];
    const unsigned char* nmask = (const unsigned char*)d_in[3];
    const float *agg_bk = F(4),  *agg_bo = F(5),  *agg_bq = F(6),  *agg_bv = F(7);
    const float *agg_wk = F(8),  *agg_wo = F(9),  *agg_wq = F(10), *agg_wv = F(11);
    const float *agg_query = F(12), *depth_emb = F(13);
    const float *edge_b1 = F(14), *edge_b2 = F(15), *edge_w1 = F(16), *edge_w2 = F(17);
    LayerP lp[L_];
    for (int l = 0; l < L_; ++l) {
        int o = 18 + 16 * l;
        lp[l] = { F(o+0), F(o+1), F(o+2), F(o+3), F(o+4), F(o+5), F(o+6), F(o+7),
                  F(o+8), F(o+9), F(o+10), F(o+11), F(o+12), F(o+13), F(o+14), F(o+15) };
    }
    const float *node_b = F(50), *node_w = F(51), *pos_emb = F(52);
    const int* parent = (const int*)d_in[53];

    // Workspace carving (floats)
    const long long SZ_POOLED = (long long)BN_ * H_;
    const long long SZ_X      = (long long)BN_ * D_;
    const long long SZ_T0     = (long long)BN_ * FF_;
    const long long SZ_SC     = (long long)B_ * NH_ * N_ * N_;
    float* ws = (float*)d_ws;
    long long off = 0;
    float* pooled = ws + off; off += SZ_POOLED;
    float* x      = ws + off; off += SZ_X;
    float* t0     = ws + off; off += SZ_T0;
    float* t1     = ws + off; off += SZ_X;
    float* t2     = ws + off; off += SZ_X;
    float* t3     = ws + off; off += SZ_X;
    float* t4     = ws + off; off += SZ_X;
    float* t5     = ws + off; off += SZ_X;
    float* t6     = ws + off; off += SZ_X;
    float* t7     = ws + off; off += SZ_X;
    float* scores = ws + off; off += SZ_SC;
    float* maskN  = ws + off; off += 352;
    float* qagg   = ws + off; off += D_;
    float* oagg   = ws + off; off += (long long)B_ * D_;
    if ((size_t)(off * sizeof(float)) > ws_size) return;   // workspace too small

    const int gridE = (int)(((long long)BN_ * D_ + 255) / 256);

    // 1) masked mean pool (BW-bound: 179 MB read ~ 8us at 23.3 TB/s)
    pool_kernel<<<BN_, 256, 0, stream>>>(states, nmask, pooled);

    // 2) x = pooled @ node_w + node_b ; += embeddings
    launch_gemm(pooled, node_w, node_b, nullptr, nullptr, 1, x,
                BN_, H_, D_, 1, 0, 0, 0, ACT_NONE, stream);
    add_emb_kernel<<<gridE, 256, 0, stream>>>(x, depth_emb, pos_emb, depth);

    // 3) edge MLP per depth level
    for (int lvl = 1; lvl < 5; ++lvl) {
        edge_mask_kernel<<<2, 256, 0, stream>>>(maskN, depth, parent, lvl);
        gather_concat_kernel<<<gridE, 256, 0, stream>>>(x, parent, t0);
        launch_gemm(t0, edge_w1, edge_b1, nullptr, nullptr, 1, t1,
                    BN_, 2 * D_, D_, 1, 0, 0, 0, ACT_GELU, stream);
        launch_gemm(t1, edge_w2, edge_b2, x, maskN, N_, x,
                    BN_, D_, D_, 1, 0, 0, 0, ACT_NONE, stream);
    }

    // 4) transformer layers
    for (int l = 0; l < L_; ++l) {
        const LayerP& P = lp[l];
        launch_gemm(x, P.wq, P.bq, nullptr, nullptr, 1, t1, BN_, D_, D_, 1, 0, 0, 0, ACT_NONE, stream);
        launch_gemm(x, P.wk, P.bk, nullptr, nullptr, 1, t2, BN_, D_, D_, 1, 0, 0, 0, ACT_NONE, stream);
        launch_gemm(x, P.wv, P.bv, nullptr, nullptr, 1, t3, BN_, D_, D_, 1, 0, 0, 0, ACT_NONE, stream);
        pack_heads_kernel<<<gridE, 256, 0, stream>>>(t1, t4, nullptr);
        pack_heads_kernel<<<gridE, 256, 0, stream>>>(t2, nullptr, t5);
        pack_heads_kernel<<<gridE, 256, 0, stream>>>(t3, t6, nullptr);
        // S = Q @ K^T  (batch = B*NH, M=N=341, K=dh)
        launch_gemm(t4, t5, nullptr, nullptr, nullptr, 1, scores,
                    N_, DH_, N_, B_ * NH_,
                    (long long)N_ * DH_, (long long)DH_ * N_, (long long)N_ * N_,
                    ACT_NONE, stream);
        softmax_kernel<<<B_ * NH_ * N_, 128, 0, stream>>>(scores, active);
        // O = P @ V  (M=341, K=341, N=dh)
        launch_gemm(scores, t6, nullptr, nullptr, nullptr, 1, t7,
                    N_, N_, DH_, B_ * NH_,
                    (long long)N_ * N_, (long long)N_ * DH_, (long long)N_ * DH_,
                    ACT_NONE, stream);
        unpack_heads_kernel<<<gridE, 256, 0, stream>>>(t7, t1);
        launch_gemm(t1, P.wo, P.bo, x, nullptr, 1, x, BN_, D_, D_, 1, 0, 0, 0, ACT_NONE, stream);
        layernorm_kernel<<<BN_, 256, 0, stream>>>(x, P.ln1_g, P.ln1_b);
        launch_gemm(x, P.ffw1, P.ffb1, nullptr, nullptr, 1, t0,
                    BN_, D_, FF_, 1, 0, 0, 0, ACT_RELU, stream);
        launch_gemm(t0, P.ffw2, P.ffb2, x, nullptr, 1, x,
                    BN_, FF_, D_, 1, 0, 0, 0, ACT_NONE, stream);
        layernorm_kernel<<<BN_, 256, 0, stream>>>(x, P.ln2_g, P.ln2_b);
    }

    // 5) aggregation attention (single query per batch)
    agg_q_kernel<<<1, 256, 0, stream>>>(agg_query, agg_wq, agg_bq, qagg);
    launch_gemm(x, agg_wk, agg_bk, nullptr, nullptr, 1, t2, BN_, D_, D_, 1, 0, 0, 0, ACT_NONE, stream);
    launch_gemm(x, agg_wv, agg_bv, nullptr, nullptr, 1, t3, BN_, D_, D_, 1, 0, 0, 0, ACT_NONE, stream);
    agg_attn_kernel<<<B_ * NH_, 256, 0, stream>>>(qagg, t2, t3, active, oagg);
    launch_gemm(oagg, agg_wo, agg_bo, nullptr, nullptr, 1, (float*)d_out,
                B_, D_, D_, 1, 0, 0, 0, ACT_NONE, stream);
}